// OuterProductMean_30760555774387
// MI455X (gfx1250) — compile-verified
//
#include <hip/hip_runtime.h>

typedef __attribute__((ext_vector_type(2))) float v2f;
typedef __attribute__((ext_vector_type(8))) float v8f;

#define NSEQ 128
#define NRES 384
#define CM   256
#define CH   32
#define CZ   128
#define NC   (NRES*CH)   // 12288
#define EPSM  1e-3f
#define LNEPS 1e-5f

// ---------------------------------------------------------------------------
// Kernel 1: fused LayerNorm + dual projection.
//   a_t[(n*32+c), s]                      : K-major rows for GEMM-1 A tiles
//   b_p[s/2][(n*32+e)*2 + (s&1)]          : s-pair-interleaved so a B-frag
//                                           {B[k,n],B[k+1,n]} is one b64
// ---------------------------------------------------------------------------
__global__ __launch_bounds__(256) void k_ln_proj(
    const float* __restrict__ m, const float* __restrict__ mask,
    const float* __restrict__ ln_scale, const float* __restrict__ ln_bias,
    const float* __restrict__ w1, const float* __restrict__ b1,
    const float* __restrict__ w2, const float* __restrict__ b2,
    float* __restrict__ a_t, float* __restrict__ b_p)
{
    const int row = blockIdx.x;            // 0..49151
    const int s = row / NRES, n = row % NRES;
    const int t = threadIdx.x;             // 0..255

    __shared__ float red[256], red2[256], lnv[256], pp[256];

    float x = m[(size_t)row*CM + t];
    red[t] = x; red2[t] = x*x;
    __syncthreads();
    for (int off = 128; off > 0; off >>= 1) {
        if (t < off) { red[t] += red[t+off]; red2[t] += red2[t+off]; }
        __syncthreads();
    }
    float mu  = red[0]  * (1.0f/CM);
    float var = red2[0] * (1.0f/CM) - mu*mu;
    float inv = rsqrtf(var + LNEPS);
    lnv[t] = (x - mu) * inv * ln_scale[t] + ln_bias[t];
    __syncthreads();

    // 64 outputs (32 for a, 32 for b); each computed by 4 threads over 64 elems
    const int o = t & 63, part = t >> 6;
    const int c = o & 31;
    const float* w = (o < CH) ? w1 : w2;
    const float* wr = w + c*CM + part*64;
    float ps = 0.f;
    #pragma unroll 8
    for (int q = 0; q < 64; ++q) ps += lnv[part*64 + q] * wr[q];
    pp[t] = ps;
    __syncthreads();
    if (t < 64) {
        float dot = pp[t] + pp[t+64] + pp[t+128] + pp[t+192];
        float mk  = mask[s*NRES + n];
        if (t < CH) {
            a_t[((size_t)n*CH + t)*NSEQ + s] = (dot + b1[t]) * mk;
        } else {
            int e = t - CH;
            b_p[(size_t)(s >> 1)*(NC*2) + (n*CH + e)*2 + (s & 1)] =
                (dot + b2[e]) * mk;
        }
    }
}

// ---------------------------------------------------------------------------
// Kernel 2: fused outer-product-mean tile.
//   Phase A: outer[(i,c),(j,e)] 128x128 via v_wmma_f32_16x16x4_f32, K=s=128,
//            operands staged by global_load_async_to_lds_b128, double-buffered.
//   Phase B: out[z, pair] = w_out[z,1024] x outer_vec[1024, 16 pairs]
// Block = 256 threads (8 waves); grid = (96,96) tiles of 4x4 (i,j) pairs.
// ---------------------------------------------------------------------------
__global__ __launch_bounds__(256) void k_outer_fused(
    const float* __restrict__ a_t, const float* __restrict__ b_p,
    const float* __restrict__ w_out, const float* __restrict__ b_out,
    const float* __restrict__ mask, float* __restrict__ out)
{
    __shared__ union SH {
        struct { float A[2][128*32]; float B[2][16*256]; } p;  // 64 KB (phase A)
        float outer[128*128];                                  // 64 KB (phase B)
    } sh;

    const int t    = threadIdx.x;
    const int lane = t & 31;
    const int wv   = t >> 5;            // wave 0..7
    const int bx   = blockIdx.x;        // i tile (4 i's)
    const int by   = blockIdx.y;        // j tile (4 j's)

    const int lm    = lane & 15;        // N / M index within 16x16 tile
    const int halfk = (lane >> 4) << 1; // K sub-offset: 0 or 2
    const int hh    = (lane >> 4) << 3; // C/D row offset: 0 or 8

    // issue one 32-deep K chunk: A 128x32, B(paired) 16x256, 8 async ops/thread
    auto issue_chunk = [&](int kc, int buf) {
        const int kb = kc * 32;
        #pragma unroll
        for (int q = 0; q < 4; ++q) {
            int f = q*256 + t;
            int r  = f >> 3, c4 = (f & 7) << 2;       // A: 128 rows x 8 f4
            unsigned la = (unsigned)(uintptr_t)(sh.p.A[buf] + r*32 + c4);
            unsigned ga = (unsigned)(((bx*128 + r)*128 + kb + c4) * 4);
            asm volatile("global_load_async_to_lds_b128 %0, %1, %2"
                         :: "v"(la), "v"(ga), "s"(a_t) : "memory");
            int sr = f >> 6, d4 = (f & 63) << 2;      // B: 16 rows x 64 f4
            unsigned lb = (unsigned)(uintptr_t)(sh.p.B[buf] + sr*256 + d4);
            unsigned gb = (unsigned)((((kb >> 1) + sr)*(NC*2) + by*256 + d4) * 4);
            asm volatile("global_load_async_to_lds_b128 %0, %1, %2"
                         :: "v"(lb), "v"(gb), "s"(b_p) : "memory");
        }
    };

    v8f acc[8];
    #pragma unroll
    for (int i = 0; i < 8; ++i) acc[i] = v8f{};

    // ---- Phase A: 128x128x128 GEMM, K in 4 chunks, async double-buffered ----
    issue_chunk(0, 0);
    #pragma unroll
    for (int kc = 0; kc < 4; ++kc) {
        const int buf = kc & 1;
        __syncthreads();                       // all waves done compute kc-1
        if (kc < 3) {
            issue_chunk(kc + 1, buf ^ 1);      // DMA next chunk under compute
            asm volatile("s_wait_asynccnt 0x8" ::: "memory");  // chunk kc landed
        } else {
            asm volatile("s_wait_asynccnt 0x0" ::: "memory");
        }
        __syncthreads();                       // chunk kc visible to all waves
        #pragma unroll
        for (int ks = 0; ks < 8; ++ks) {
            v2f af = *(const v2f*)(sh.p.A[buf] + (wv*16 + lm)*32 + ks*4 + halfk);
            const float* bp = sh.p.B[buf] + (ks*2 + (halfk >> 1))*256 + lm*2;
            #pragma unroll
            for (int nt = 0; nt < 8; ++nt) {
                v2f bf = *(const v2f*)(bp + nt*32);      // one ds_load_b64
                acc[nt] = __builtin_amdgcn_wmma_f32_16x16x4_f32(
                    false, af, false, bf, (short)0, acc[nt], false, false);
            }
        }
    }

    // ---- spill outer block to LDS (union reuse) ----
    __syncthreads();
    #pragma unroll
    for (int nt = 0; nt < 8; ++nt)
        #pragma unroll
        for (int r = 0; r < 8; ++r)
            sh.outer[(wv*16 + r + hh)*128 + nt*16 + lm] = acc[nt][r];
    __syncthreads();

    // ---- Phase B: out[z, pair], M=128 (wave->16 z's), N=16 pairs, K=1024 ----
    const int it = lm >> 2, jt = lm & 3;     // pair -> (i,j) within tile
    v8f acc2 = v8f{};
    const int zbase = wv*16;
    #pragma unroll 4
    for (int kk = 0; kk < CH*CH; kk += 4) {
        int k = kk + halfk;
        v2f a2 = *(const v2f*)(w_out + (size_t)(zbase + lm)*(CH*CH) + k);
        int cc = k >> 5, ee = k & 31;
        v2f b2 = *(const v2f*)(sh.outer + (it*32 + cc)*128 + jt*32 + ee);
        acc2 = __builtin_amdgcn_wmma_f32_16x16x4_f32(
            false, a2, false, b2, (short)0, acc2, false, false);
    }

    // ---- epilogue: bias + mask-norm divide ----
    const int gi = bx*4 + it, gj = by*4 + jt;
    float nrm = EPSM;
    for (int s = 0; s < NSEQ; ++s)
        nrm += mask[s*NRES + gi] * mask[s*NRES + gj];
    float invn = 1.0f / nrm;
    #pragma unroll
    for (int r = 0; r < 8; ++r) {
        int z = zbase + r + hh;
        out[((size_t)gi*NRES + gj)*CZ + z] = (acc2[r] + b_out[z]) * invn;
    }
}

// ---------------------------------------------------------------------------
extern "C" void kernel_launch(void* const* d_in, const int* in_sizes, int n_in,
                              void* d_out, int out_size, void* d_ws, size_t ws_size,
                              hipStream_t stream) {
    const float* m        = (const float*)d_in[0];
    const float* mask     = (const float*)d_in[1];
    const float* ln_scale = (const float*)d_in[2];
    const float* ln_bias  = (const float*)d_in[3];
    const float* w1       = (const float*)d_in[4];
    const float* b1       = (const float*)d_in[5];
    const float* w2       = (const float*)d_in[6];
    const float* b2       = (const float*)d_in[7];
    const float* w_out    = (const float*)d_in[8];
    const float* b_out    = (const float*)d_in[9];
    float* out = (float*)d_out;

    float* a_t = (float*)d_ws;                    // [12288, 128]       6 MB
    float* b_p = a_t + (size_t)NC*NSEQ;           // [64, 12288*2]      6 MB

    k_ln_proj<<<dim3(NSEQ*NRES), dim3(256), 0, stream>>>(
        m, mask, ln_scale, ln_bias, w1, b1, w2, b2, a_t, b_p);
    k_outer_fused<<<dim3(NRES/4, NRES/4), dim3(256), 0, stream>>>(
        a_t, b_p, w_out, b_out, mask, out);
}